// UniPhyTransformerBlock_53687091200318
// MI455X (gfx1250) — compile-verified
//
#include <hip/hip_runtime.h>
#include <math.h>

typedef __attribute__((ext_vector_type(2))) float v2f;
typedef __attribute__((ext_vector_type(8))) float v8f;

#define DEV __device__ __forceinline__

constexpr int  Bd = 2, Td = 16, Dd = 64, Hd = 64, Wd = 64;
constexpr int  Pd = Hd * Wd;          // 4096 pixels per (bt,d) plane
constexpr int  BT = Bd * Td;          // 32
constexpr long DP = (long)Dd * Pd;    // 262144 floats per bt
constexpr long Ub = (long)BT * DP;    // 8388608 floats per full buffer (32 MiB)

// workspace float offsets: 10 big slots + small region
constexpr long SM     = 10 * Ub;
constexpr long OF_Fr  = SM + 0;        // DFT matrix real (64x64)
constexpr long OF_Fi  = SM + 4096;     // DFT matrix imag
constexpr long OF_FIr = SM + 8192;     // conj(F)/64 real (one inverse-axis step)
constexpr long OF_FIi = SM + 12288;    // conj(F)/64 imag
constexpr long OF_Er  = SM + 16384;    // evo real  (BT*D = 2048)
constexpr long OF_Ei  = SM + 18432;    // evo imag

DEV v8f wmma4(v2f a, v2f b, v8f c) {
  // D = A(16x4 f32) * B(4x16 f32) + C(16x16 f32)
  return __builtin_amdgcn_wmma_f32_16x16x4_f32(false, a, false, b, (short)0, c,
                                               false, false);
}

DEV v8f vz() {
  v8f z;
#pragma unroll
  for (int j = 0; j < 8; ++j) z[j] = 0.f;
  return z;
}

// ---------------------------------------------------------------------------
// Precompute: DFT matrices F (symmetric) and Fi = conj(F)/64; evo = exp(lam*dt)
// ---------------------------------------------------------------------------
__global__ __launch_bounds__(256) void k_pre(const float* __restrict__ dt,
                                             const float* __restrict__ lre,
                                             const float* __restrict__ lim,
                                             float* __restrict__ ws) {
  int g = blockIdx.x * 256 + threadIdx.x;
  if (g < 4096) {
    int h = g >> 6, w = g & 63;
    int m = (h * w) & 63;
    float ang = -6.283185307179586f * (float)m * (1.f / 64.f);
    float c = cosf(ang), s = sinf(ang);
    ws[OF_Fr + g] = c;
    ws[OF_Fi + g] = s;
    ws[OF_FIr + g] = c * (1.f / 64.f);
    ws[OF_FIi + g] = -s * (1.f / 64.f);
  } else if (g < 4096 + 2048) {
    int idx = g - 4096;
    int bt = idx >> 6, d = idx & 63;
    float dtv = dt[bt];
    float x = lre[d];
    float sp = (x > 20.f) ? x : log1pf(expf(x));   // softplus
    float mag = expf(-sp * dtv);
    float th = lim[d] * dtv;
    ws[OF_Er + idx] = mag * cosf(th);
    ws[OF_Ei + idx] = mag * sinf(th);
  }
}

// ---------------------------------------------------------------------------
// Channel LayerNorm over concat(real,imag) (128 ch), one thread per token.
// Layout: arr[bt][d][p], token = (bt, p). Coalesced over p.
// ---------------------------------------------------------------------------
__global__ __launch_bounds__(256) void k_ln(const float* __restrict__ xr,
                                            const float* __restrict__ xi,
                                            const float* __restrict__ w,
                                            const float* __restrict__ b,
                                            float* __restrict__ yr,
                                            float* __restrict__ yi) {
  int p = blockIdx.x * 256 + threadIdx.x;
  long base = (long)blockIdx.y * DP + p;
  float s = 0.f, ss = 0.f;
  for (int d = 0; d < Dd; ++d) {
    float a = xr[base + (long)d * Pd];
    float c = xi[base + (long)d * Pd];
    s += a + c;
    ss += a * a + c * c;
  }
  float mu = s * (1.f / 128.f);
  float var = ss * (1.f / 128.f) - mu * mu;
  float inv = rsqrtf(var + 1e-5f);
  for (int d = 0; d < Dd; ++d) {
    long o = base + (long)d * Pd;
    yr[o] = (xr[o] - mu) * inv * w[d] + b[d];
    yi[o] = (xi[o] - mu) * inv * w[64 + d] + b[64 + d];
  }
}

// ---------------------------------------------------------------------------
// Clifford 3x3 complex conv as 9 tap-GEMMs (WMMA). Block: 16 pixels (one row
// segment) x all 64 out channels (4 waves). Haloed LDS tile, zero-padded.
// ---------------------------------------------------------------------------
__global__ __launch_bounds__(128) void k_conv(
    const float* __restrict__ xr, const float* __restrict__ xi,
    const float* __restrict__ wr, const float* __restrict__ wi,
    const float* __restrict__ br, const float* __restrict__ bi,
    float* __restrict__ outr, float* __restrict__ outi) {
  __shared__ float HalR[3 * 64 * 18];
  __shared__ float HalI[3 * 64 * 18];
  int bt = blockIdx.y;
  int y = blockIdx.x >> 2;
  int x0 = (blockIdx.x & 3) * 16;
  long ibase = (long)bt * DP;
  int tid = threadIdx.x;
  for (int t = tid; t < 3 * 64 * 18; t += 128) {
    int r = t / (64 * 18);
    int rem = t % (64 * 18);
    int d = rem / 18, col = rem % 18;
    int gy = y + r - 1, gx = x0 + col - 1;
    bool ok = (gy >= 0 && gy < 64 && gx >= 0 && gx < 64);
    long gi = ibase + (long)d * Pd + gy * 64 + gx;
    HalR[t] = ok ? xr[gi] : 0.f;
    HalI[t] = ok ? xi[gi] : 0.f;
  }
  __syncthreads();
  int lane = tid & 31, wave = tid >> 5;
  int mcol = lane & 15, hi2 = (lane >> 4) * 2;
  int n = wave * 16 + mcol;  // out channel
  v8f RR = vz(), II = vz(), RI = vz(), IR = vz();
  for (int tap = 0; tap < 9; ++tap) {
    int ky = tap / 3, kx = tap % 3;
    int hb = ky * (64 * 18);
    int wb = n * 576 + ky * 3 + kx;  // cw[n][k][ky][kx], k-stride 9
    for (int k0 = 0; k0 < 64; k0 += 4) {
      int k = k0 + hi2;
      v2f ar, ai, cbr, cbi;
      ar.x = HalR[hb + k * 18 + mcol + kx];
      ar.y = HalR[hb + (k + 1) * 18 + mcol + kx];
      ai.x = HalI[hb + k * 18 + mcol + kx];
      ai.y = HalI[hb + (k + 1) * 18 + mcol + kx];
      cbr.x = wr[wb + k * 9];
      cbr.y = wr[wb + (k + 1) * 9];
      cbi.x = wi[wb + k * 9];
      cbi.y = wi[wb + (k + 1) * 9];
      RR = wmma4(ar, cbr, RR);
      II = wmma4(ai, cbi, II);
      RI = wmma4(ar, cbi, RI);
      IR = wmma4(ai, cbr, IR);
    }
  }
  float bre = br[n], bim = bi[n];
  long ob = ibase + (long)n * Pd + y * 64 + x0 + (lane >> 4) * 8;
#pragma unroll
  for (int j = 0; j < 8; ++j) {
    outr[ob + j] = RR[j] - II[j] + bre;
    outi[ob + j] = RI[j] + IR[j] + bim;
  }
}

// ---------------------------------------------------------------------------
// DFT along last axis (w): C(row,n) = sum_k A(row,k) * F(k,n). Rows contiguous.
// blend=1: epilogue x1 = g*cliff + (1-g)*C + x_input (same flat indexing).
// ---------------------------------------------------------------------------
__global__ __launch_bounds__(128) void k_fft_w(
    const float* __restrict__ inr, const float* __restrict__ ini,
    const float* __restrict__ Fr, const float* __restrict__ Fi,
    float* __restrict__ outr, float* __restrict__ outi, int blend,
    const float* __restrict__ clr, const float* __restrict__ cli,
    const float* __restrict__ x0r, const float* __restrict__ x0i,
    const float* __restrict__ gate) {
  __shared__ float AsR[16 * 68];
  __shared__ float AsI[16 * 68];
  long r0 = (long)blockIdx.x * 16;
  int tid = threadIdx.x;
  for (int t = tid; t < 16 * 16; t += 128) {
    int row = t >> 4, q = (t & 15) * 4;
    *(float4*)(AsR + row * 68 + q) = *(const float4*)(inr + (r0 + row) * 64 + q);
    *(float4*)(AsI + row * 68 + q) = *(const float4*)(ini + (r0 + row) * 64 + q);
  }
  __syncthreads();
  int lane = tid & 31, wave = tid >> 5;
  int mcol = lane & 15, hi2 = (lane >> 4) * 2;
  int n = wave * 16 + mcol;
  v8f RR = vz(), II = vz(), RI = vz(), IR = vz();
  for (int k0 = 0; k0 < 64; k0 += 4) {
    int k = k0 + hi2;
    v2f ar, ai, br, bi;
    ar.x = AsR[mcol * 68 + k];
    ar.y = AsR[mcol * 68 + k + 1];
    ai.x = AsI[mcol * 68 + k];
    ai.y = AsI[mcol * 68 + k + 1];
    br.x = Fr[k * 64 + n];
    br.y = Fr[(k + 1) * 64 + n];
    bi.x = Fi[k * 64 + n];
    bi.y = Fi[(k + 1) * 64 + n];
    RR = wmma4(ar, br, RR);
    II = wmma4(ai, bi, II);
    RI = wmma4(ar, bi, RI);
    IR = wmma4(ai, br, IR);
  }
  long ob = (r0 + (lane >> 4) * 8) * 64 + n;
  if (!blend) {
#pragma unroll
    for (int j = 0; j < 8; ++j) {
      outr[ob + (long)j * 64] = RR[j] - II[j];
      outi[ob + (long)j * 64] = RI[j] + IR[j];
    }
  } else {
    float g = gate[0];
#pragma unroll
    for (int j = 0; j < 8; ++j) {
      long o = ob + (long)j * 64;
      outr[o] = g * clr[o] + (1.f - g) * (RR[j] - II[j]) + x0r[o];
      outi[o] = g * cli[o] + (1.f - g) * (RI[j] + IR[j]) + x0i[o];
    }
  }
}

// ---------------------------------------------------------------------------
// DFT along h axis per (bt,d) image: C(h',w) = sum_h F(h',h) * X(h,w).
// usespec=1: multiply result by wspec[d,h',w] (complex).
// ---------------------------------------------------------------------------
__global__ __launch_bounds__(128) void k_fft_h(
    const float* __restrict__ inr, const float* __restrict__ ini,
    const float* __restrict__ Fr, const float* __restrict__ Fi,
    float* __restrict__ outr, float* __restrict__ outi, int usespec,
    const float* __restrict__ swr, const float* __restrict__ swi) {
  __shared__ float BsR[64 * 16];
  __shared__ float BsI[64 * 16];
  int img = blockIdx.y;  // bt*64 + d
  int w0 = blockIdx.x * 16;
  long base = (long)img * Pd;
  int tid = threadIdx.x;
  for (int t = tid; t < 64 * 16; t += 128) {
    int h = t >> 4, c = t & 15;
    BsR[t] = inr[base + h * 64 + w0 + c];
    BsI[t] = ini[base + h * 64 + w0 + c];
  }
  __syncthreads();
  int lane = tid & 31, wave = tid >> 5;
  int mcol = lane & 15, hi2 = (lane >> 4) * 2;
  int m0 = wave * 16;  // h' tile
  v8f RR = vz(), II = vz(), RI = vz(), IR = vz();
  for (int k0 = 0; k0 < 64; k0 += 4) {
    int k = k0 + hi2;
    v2f ar, ai, br, bi;
    ar.x = Fr[(m0 + mcol) * 64 + k];
    ar.y = Fr[(m0 + mcol) * 64 + k + 1];
    ai.x = Fi[(m0 + mcol) * 64 + k];
    ai.y = Fi[(m0 + mcol) * 64 + k + 1];
    br.x = BsR[k * 16 + mcol];
    br.y = BsR[(k + 1) * 16 + mcol];
    bi.x = BsI[k * 16 + mcol];
    bi.y = BsI[(k + 1) * 16 + mcol];
    RR = wmma4(ar, br, RR);
    II = wmma4(ai, bi, II);
    RI = wmma4(ar, bi, RI);
    IR = wmma4(ai, br, IR);
  }
  int rb = m0 + (lane >> 4) * 8;
  if (usespec) {
    int d = img & 63;
#pragma unroll
    for (int j = 0; j < 8; ++j) {
      float cr = RR[j] - II[j], ci = RI[j] + IR[j];
      long si = (long)d * Pd + (long)(rb + j) * 64 + w0 + mcol;
      float wre = swr[si], wim = swi[si];
      long o = base + (long)(rb + j) * 64 + w0 + mcol;
      outr[o] = cr * wre - ci * wim;
      outi[o] = cr * wim + ci * wre;
    }
  } else {
#pragma unroll
    for (int j = 0; j < 8; ++j) {
      long o = base + (long)(rb + j) * 64 + w0 + mcol;
      outr[o] = RR[j] - II[j];
      outi[o] = RI[j] + IR[j];
    }
  }
}

// ---------------------------------------------------------------------------
// Complex GEMM over channels: Y(tok,n) = sum_d X(tok,d) * W(n,d)  (W accessed
// as its transpose, i.e. x @ W^T). Channel-layout in/out. addres: +residual.
// ---------------------------------------------------------------------------
__global__ __launch_bounds__(128) void k_cgemm_chan(
    const float* __restrict__ xr, const float* __restrict__ xi,
    const float* __restrict__ Wr, const float* __restrict__ Wi,
    float* __restrict__ yr, float* __restrict__ yi, int addres,
    const float* __restrict__ rr, const float* __restrict__ ri) {
  __shared__ float AsR[64 * 16];
  __shared__ float AsI[64 * 16];
  int bt = blockIdx.y;
  int p0 = blockIdx.x * 16;
  long base = (long)bt * DP;
  int tid = threadIdx.x;
  for (int t = tid; t < 64 * 4; t += 128) {
    int d = t >> 2, q = (t & 3) * 4;
    *(float4*)(AsR + d * 16 + q) =
        *(const float4*)(xr + base + (long)d * Pd + p0 + q);
    *(float4*)(AsI + d * 16 + q) =
        *(const float4*)(xi + base + (long)d * Pd + p0 + q);
  }
  __syncthreads();
  int lane = tid & 31, wave = tid >> 5;
  int mcol = lane & 15, hi2 = (lane >> 4) * 2;
  int n = wave * 16 + mcol;
  v8f RR = vz(), II = vz(), RI = vz(), IR = vz();
  for (int k0 = 0; k0 < 64; k0 += 4) {
    int k = k0 + hi2;
    v2f ar, ai, br, bi;
    ar.x = AsR[k * 16 + mcol];
    ar.y = AsR[(k + 1) * 16 + mcol];
    ai.x = AsI[k * 16 + mcol];
    ai.y = AsI[(k + 1) * 16 + mcol];
    br.x = Wr[n * 64 + k];
    br.y = Wr[n * 64 + k + 1];
    bi.x = Wi[n * 64 + k];
    bi.y = Wi[n * 64 + k + 1];
    RR = wmma4(ar, br, RR);
    II = wmma4(ai, bi, II);
    RI = wmma4(ar, bi, RI);
    IR = wmma4(ai, br, IR);
  }
  long ob = base + (long)n * Pd + p0 + (lane >> 4) * 8;
  if (addres) {
#pragma unroll
    for (int j = 0; j < 8; ++j) {
      yr[ob + j] = RR[j] - II[j] + rr[ob + j];
      yi[ob + j] = RI[j] + IR[j] + ri[ob + j];
    }
  } else {
#pragma unroll
    for (int j = 0; j < 8; ++j) {
      yr[ob + j] = RR[j] - II[j];
      yi[ob + j] = RI[j] + IR[j];
    }
  }
}

// ---------------------------------------------------------------------------
// Diagonal temporal scan (T=16), in place: h_t = evo_t * h_{t-1} + x_t.
// One thread per (b,d,p).
// ---------------------------------------------------------------------------
__global__ __launch_bounds__(256) void k_scan(float* __restrict__ hr_,
                                              float* __restrict__ hi_,
                                              const float* __restrict__ er,
                                              const float* __restrict__ ei) {
  long g = (long)blockIdx.x * 256 + threadIdx.x;  // over B*D*P
  int b = (int)(g / DP);
  long rem = g % DP;  // d*P + p
  int d = (int)(rem / Pd);
  float hr = 0.f, hi = 0.f;
  for (int t = 0; t < 16; ++t) {
    int bt = b * 16 + t;
    float evr = er[bt * 64 + d], evi = ei[bt * 64 + d];
    long idx = (long)bt * DP + rem;
    float xr = hr_[idx], xi = hi_[idx];
    float nr = evr * hr - evi * hi + xr;
    float ni = evr * hi + evi * hr + xi;
    hr = nr;
    hi = ni;
    hr_[idx] = hr;
    hi_[idx] = hi;
  }
}

// ---------------------------------------------------------------------------
// Fused MLP: 128 -> 256 (GELU exact) -> 128, + residual, write interleaved out.
// Block: 16 tokens, 8 waves. Hidden kept in LDS.
// ---------------------------------------------------------------------------
__global__ __launch_bounds__(256) void k_mlp(
    const float* __restrict__ mr, const float* __restrict__ mi,
    const float* __restrict__ w1, const float* __restrict__ b1,
    const float* __restrict__ w2, const float* __restrict__ b2,
    const float* __restrict__ rr, const float* __restrict__ ri,
    float* __restrict__ out) {
  __shared__ float As[128 * 16];
  __shared__ float Hs[256 * 16];
  int bt = blockIdx.y;
  int p0 = blockIdx.x * 16;
  long base = (long)bt * DP;
  int tid = threadIdx.x;
  for (int t = tid; t < 64 * 4; t += 256) {
    int d = t >> 2, q = (t & 3) * 4;
    *(float4*)(As + d * 16 + q) =
        *(const float4*)(mr + base + (long)d * Pd + p0 + q);
    *(float4*)(As + (64 + d) * 16 + q) =
        *(const float4*)(mi + base + (long)d * Pd + p0 + q);
  }
  __syncthreads();
  int lane = tid & 31, wave = tid >> 5;
  int mcol = lane & 15, hi2 = (lane >> 4) * 2;
  int mrow = (lane >> 4) * 8;
  // stage 1: hidden = gelu(z @ w1 + b1), 2 n-tiles per wave
  for (int s = 0; s < 2; ++s) {
    int n = (wave * 2 + s) * 16 + mcol;
    v8f acc = vz();
    for (int k0 = 0; k0 < 128; k0 += 4) {
      int k = k0 + hi2;
      v2f a, bv;
      a.x = As[k * 16 + mcol];
      a.y = As[(k + 1) * 16 + mcol];
      bv.x = w1[k * 256 + n];
      bv.y = w1[(k + 1) * 256 + n];
      acc = wmma4(a, bv, acc);
    }
    float bb = b1[n];
#pragma unroll
    for (int j = 0; j < 8; ++j) {
      float h = acc[j] + bb;
      h = 0.5f * h * (1.f + erff(h * 0.70710678118654752f));
      Hs[n * 16 + mrow + j] = h;
    }
  }
  __syncthreads();
  // stage 2: out = hidden @ w2 + b2 + resid
  {
    int n = wave * 16 + mcol;
    v8f acc = vz();
    for (int k0 = 0; k0 < 256; k0 += 4) {
      int k = k0 + hi2;
      v2f a, bv;
      a.x = Hs[k * 16 + mcol];
      a.y = Hs[(k + 1) * 16 + mcol];
      bv.x = w2[k * 128 + n];
      bv.y = w2[(k + 1) * 128 + n];
      acc = wmma4(a, bv, acc);
    }
    float bb = b2[n];
    int d = n & 63;
    int im = n >> 6;
    const float* res = im ? ri : rr;
    long ob = base + (long)d * Pd + p0 + mrow;
#pragma unroll
    for (int j = 0; j < 8; ++j) {
      float v = acc[j] + bb + res[ob + j];
      out[(ob + j) * 2 + im] = v;
    }
  }
}

// ---------------------------------------------------------------------------
extern "C" void kernel_launch(void* const* d_in, const int* in_sizes, int n_in,
                              void* d_out, int out_size, void* d_ws,
                              size_t ws_size, hipStream_t stream) {
  (void)in_sizes; (void)n_in; (void)out_size; (void)ws_size;
  const float* x_r    = (const float*)d_in[0];
  const float* x_i    = (const float*)d_in[1];
  const float* dt     = (const float*)d_in[2];
  const float* ln_s_w = (const float*)d_in[3];
  const float* ln_s_b = (const float*)d_in[4];
  const float* ln_t_w = (const float*)d_in[5];
  const float* ln_t_b = (const float*)d_in[6];
  const float* ln_m_w = (const float*)d_in[7];
  const float* ln_m_b = (const float*)d_in[8];
  const float* cw_r   = (const float*)d_in[9];
  const float* cw_i   = (const float*)d_in[10];
  const float* cb_r   = (const float*)d_in[11];
  const float* cb_i   = (const float*)d_in[12];
  const float* spec_wr= (const float*)d_in[13];
  const float* spec_wi= (const float*)d_in[14];
  const float* gate   = (const float*)d_in[15];
  const float* lam_re = (const float*)d_in[16];
  const float* lam_im = (const float*)d_in[17];
  const float* V_re   = (const float*)d_in[18];
  const float* V_im   = (const float*)d_in[19];
  const float* Vi_re  = (const float*)d_in[20];
  const float* Vi_im  = (const float*)d_in[21];
  const float* mlp_w1 = (const float*)d_in[22];
  const float* mlp_b1 = (const float*)d_in[23];
  const float* mlp_w2 = (const float*)d_in[24];
  const float* mlp_b2 = (const float*)d_in[25];

  float* ws  = (float*)d_ws;
  float* out = (float*)d_out;
  float* s0 = ws + 0 * Ub;  float* s1 = ws + 1 * Ub;   // LN outputs
  float* s2 = ws + 2 * Ub;  float* s3 = ws + 3 * Ub;   // cliff / x_eig
  float* s4 = ws + 4 * Ub;  float* s5 = ws + 5 * Ub;   // fft ping
  float* s6 = ws + 6 * Ub;  float* s7 = ws + 7 * Ub;   // fft pong / x2
  float* s8 = ws + 8 * Ub;  float* s9 = ws + 9 * Ub;   // x1
  float* Fr = ws + OF_Fr;   float* Fi = ws + OF_Fi;
  float* FIr = ws + OF_FIr; float* FIi = ws + OF_FIi;
  float* Er = ws + OF_Er;   float* Ei = ws + OF_Ei;

  // precompute DFT matrices + evo factors
  k_pre<<<24, 256, 0, stream>>>(dt, lam_re, lam_im, ws);

  // ---- spatial mixing ----
  k_ln<<<dim3(16, 32), 256, 0, stream>>>(x_r, x_i, ln_s_w, ln_s_b, s0, s1);
  k_conv<<<dim3(256, 32), 128, 0, stream>>>(s0, s1, cw_r, cw_i, cb_r, cb_i, s2, s3);
  k_fft_w<<<8192, 128, 0, stream>>>(s0, s1, Fr, Fi, s4, s5, 0,
                                    nullptr, nullptr, nullptr, nullptr, nullptr);
  k_fft_h<<<dim3(4, 2048), 128, 0, stream>>>(s4, s5, Fr, Fi, s6, s7, 1,
                                             spec_wr, spec_wi);
  k_fft_h<<<dim3(4, 2048), 128, 0, stream>>>(s6, s7, FIr, FIi, s4, s5, 0,
                                             nullptr, nullptr);
  k_fft_w<<<8192, 128, 0, stream>>>(s4, s5, FIr, FIi, s8, s9, 1,
                                    s2, s3, x_r, x_i, gate);  // -> x1

  // ---- temporal mixing ----
  k_ln<<<dim3(16, 32), 256, 0, stream>>>(s8, s9, ln_t_w, ln_t_b, s0, s1);
  k_cgemm_chan<<<dim3(256, 32), 128, 0, stream>>>(s0, s1, Vi_re, Vi_im, s2, s3,
                                                  0, nullptr, nullptr);
  k_scan<<<2048, 256, 0, stream>>>(s2, s3, Er, Ei);
  k_cgemm_chan<<<dim3(256, 32), 128, 0, stream>>>(s2, s3, V_re, V_im, s6, s7,
                                                  1, s8, s9);  // -> x2

  // ---- complex MLP ----
  k_ln<<<dim3(16, 32), 256, 0, stream>>>(s6, s7, ln_m_w, ln_m_b, s0, s1);
  k_mlp<<<dim3(256, 32), 256, 0, stream>>>(s0, s1, mlp_w1, mlp_b1, mlp_w2,
                                           mlp_b2, s6, s7, out);
}